// HeatmapCELoss_75720273428986
// MI455X (gfx1250) — compile-verified
//
#include <hip/hip_runtime.h>
#include <hip/hip_bf16.h>

typedef __attribute__((ext_vector_type(2))) float v2f;
typedef __attribute__((ext_vector_type(8))) float v8f;

#define B_DIM 64
#define V_DIM 8
#define J_DIM 17
#define H_DIM 256
#define N_TOTAL (B_DIM * V_DIM * J_DIM)   // 8704
#define NTHREADS 512                       // 8704 / 512 = 17 exactly, no tail
#define ITERS (N_TOTAL / NTHREADS)         // 17

// Single-workgroup gather + -log + deterministic reduction.
// Final 32-lane reduction uses V_WMMA_F32_16X16X4_F32 as an exact f32 adder tree:
//   A(16x4): lane partials at K=0 (lanes 0-15) and K=2 (lanes 16-31), zeros at K=1,3
//   B(4x16): all ones (lane layout irrelevant for a constant matrix)
//   => D[m][n] = p[m] + p[m+16]; per-lane sum of 8 D VGPRs + shfl_xor(16) = total.
__global__ __launch_bounds__(NTHREADS) void heatmap_ce_loss_kernel(
    const float* __restrict__ kp,   // (B, V, J, 2) float32
    const float* __restrict__ hm,   // (B, V, 1, H, H) float32
    float* __restrict__ out)        // scalar
{
    __shared__ float partial[NTHREADS];
    const int tid = threadIdx.x;

    float acc = 0.0f;
    #pragma unroll
    for (int it = 0; it < ITERS; ++it) {
        const int e = tid + it * NTHREADS;          // e = (b*V + v)*J + j
        const float vx = kp[2 * e + 0];
        const float vy = kp[2 * e + 1];
        int xi = (int)ceilf(vx);
        int yi = (int)ceilf(vy);
        xi = min(max(xi, 0), H_DIM - 1);
        yi = min(max(yi, 0), H_DIM - 1);
        const int bv = e / J_DIM;                   // b*V + v
        const float val =
            hm[(size_t)bv * (size_t)(H_DIM * H_DIM) + (size_t)(yi * H_DIM + xi)];
        acc += -logf(val);
    }

    partial[tid] = acc;
    __syncthreads();

    if (tid < 32) {                                  // wave 0, EXEC all ones (WMMA requirement)
        float s = 0.0f;
        #pragma unroll
        for (int w = 0; w < NTHREADS / 32; ++w)
            s += partial[tid + w * 32];

#if defined(__gfx1250__) && __has_builtin(__builtin_amdgcn_wmma_f32_16x16x4_f32)
        v2f a;
        a[0] = s;      // lanes 0-15: A[m][K=0]; lanes 16-31: A[m][K=2]
        a[1] = 0.0f;   // K=1 / K=3 -> exact zero contribution
        v2f b;
        b[0] = 1.0f;   // B(4x16) == all ones
        b[1] = 1.0f;
        v8f c = {};
        c = __builtin_amdgcn_wmma_f32_16x16x4_f32(
            /*neg_a=*/false, a, /*neg_b=*/false, b,
            /*c_mod=*/(short)0, c, /*reuse_a=*/false, /*reuse_b=*/false);
        // lane n (0-15) holds D[0..7][n]; lane n+16 holds D[8..15][n]
        float rs = c[0] + c[1] + c[2] + c[3] + c[4] + c[5] + c[6] + c[7];
        float total = rs + __shfl_xor(rs, 16, 32);   // every lane: full 32-lane sum
        if (tid == 0)
            out[0] = total * (1.0f / (float)N_TOTAL);
#else
        #pragma unroll
        for (int off = 16; off > 0; off >>= 1)
            s += __shfl_xor(s, off, 32);
        if (tid == 0)
            out[0] = s * (1.0f / (float)N_TOTAL);
#endif
    }
}

extern "C" void kernel_launch(void* const* d_in, const int* in_sizes, int n_in,
                              void* d_out, int out_size, void* d_ws, size_t ws_size,
                              hipStream_t stream) {
    (void)in_sizes; (void)n_in; (void)d_ws; (void)ws_size; (void)out_size;
    const float* kp = (const float*)d_in[0];   // keypoints_gt (64,8,17,2) f32
    const float* hm = (const float*)d_in[1];   // heatmap (64,8,1,256,256) f32
    float* out = (float*)d_out;                // scalar f32
    heatmap_ce_loss_kernel<<<1, NTHREADS, 0, stream>>>(kp, hm, out);
}